// MolecularGraphNeuralNetwork_89172110999867
// MI455X (gfx1250) — compile-verified
//
#include <hip/hip_runtime.h>
#include <math.h>

#define NN 50000
#define EE 800000
#define DD 128
#define BB 512
#define ALPHA 0.01f
#define EPSBN 1e-5f

typedef __attribute__((ext_vector_type(2))) float v2f;
typedef __attribute__((ext_vector_type(8))) float v8f;

__device__ __forceinline__ float leakyf(float v) { return v > 0.f ? v : ALPHA * v; }

__device__ __forceinline__ void atomicMaxFloat(float* addr, float val) {
    if (val >= 0.f) atomicMax((int*)addr, __float_as_int(val));
    else            atomicMin((unsigned int*)addr, __float_as_uint(val));
}

__global__ void fill_kernel(float* __restrict__ p, long n, float v) {
    long i = (long)blockIdx.x * blockDim.x + threadIdx.x;
    long stride = (long)gridDim.x * blockDim.x;
    for (; i < n; i += stride) p[i] = v;
}

__global__ void deg_kernel(float* __restrict__ deg, const int* __restrict__ dst, int e) {
    int i = blockIdx.x * blockDim.x + threadIdx.x;
    if (i < e) atomicAdd(&deg[dst[i]], 1.0f);
}

__global__ void rs_kernel(float* __restrict__ rs, const float* __restrict__ deg, int n) {
    int i = blockIdx.x * blockDim.x + threadIdx.x;
    if (i < n) rs[i] = 1.0f / fmaxf(deg[i], 1.0f);
}

// out[n,128] = x[n,9] @ W0[9,128]  (K=9: scalar path; WMMA not worth it)
__global__ void lin0_kernel(const float* __restrict__ x, const float* __restrict__ W0,
                            float* __restrict__ out, int n) {
    int tid = blockIdx.x * blockDim.x + threadIdx.x;
    if (tid >= n * DD) return;
    int r = tid >> 7, d = tid & 127;
    float s = 0.f;
#pragma unroll
    for (int k = 0; k < 9; ++k) s += x[r * 9 + k] * W0[k * DD + d];
    out[tid] = s;
}

// mode 0: out[dst] += h[src]; mode 1: GCN  out[dst] += h[src]*rsqrt(deg[s]+1)*rsqrt(deg[d]+1)
__global__ void scatter_kernel(float* __restrict__ out, const float* __restrict__ h,
                               const int* __restrict__ srcIdx, const int* __restrict__ dstIdx,
                               const float* __restrict__ deg, int e, int mode) {
    long tid = (long)blockIdx.x * blockDim.x + threadIdx.x;
    if (tid >= (long)e * 32) return;
    int ed = (int)(tid >> 5);
    int d0 = ((int)tid & 31) * 4;
    int s = srcIdx[ed], d = dstIdx[ed];
    float c = 1.f;
    if (mode == 1) c = rsqrtf(deg[s] + 1.f) * rsqrtf(deg[d] + 1.f);
    const float* sp = h + (size_t)s * DD + d0;
    float* op = out + (size_t)d * DD + d0;
#pragma unroll
    for (int i = 0; i < 4; ++i) atomicAdd(op + i, sp[i] * c);
}

// out = leaky(scat + hlin * 1/(deg+1) + bias)
__global__ void gcn_finish_kernel(const float* __restrict__ scat, const float* __restrict__ hlin,
                                  const float* __restrict__ deg, const float* __restrict__ bias,
                                  float* __restrict__ out, int n) {
    int tid = blockIdx.x * blockDim.x + threadIdx.x;
    if (tid >= n * DD) return;
    int r = tid >> 7, d = tid & 127;
    float di = 1.f / (deg[r] + 1.f);
    out[tid] = leakyf(scat[tid] + hlin[tid] * di + bias[d]);
}

__global__ void bn_stats_kernel(const float* __restrict__ h, float* __restrict__ stats, int nrows) {
    int c = threadIdx.x & 127;
    int r = blockIdx.x * 2 + (threadIdx.x >> 7);
    float s = 0.f, sq = 0.f;
    for (; r < nrows; r += gridDim.x * 2) {
        float v = h[(size_t)r * DD + c];
        s += v; sq += v * v;
    }
    atomicAdd(&stats[c], s);
    atomicAdd(&stats[DD + c], sq);
}

__global__ void bn_apply_kernel(float* __restrict__ h, const float* __restrict__ stats,
                                const float* __restrict__ gam, const float* __restrict__ bet, int nrows) {
    int tid = blockIdx.x * blockDim.x + threadIdx.x;
    if (tid >= nrows * DD) return;
    int c = tid & 127;
    float inv_n = 1.0f / (float)nrows;
    float m = stats[c] * inv_n;
    float var = stats[DD + c] * inv_n - m * m;
    h[tid] = gam[c] * (h[tid] - m) * rsqrtf(var + EPSBN) + bet[c];
}

// C[nrows x 128] = act( (A1 .* rowscale) @ W1 [+ A2 @ W2] + bias )
// W1: [K x 128]; when DUAL, K == 128 and W2: [128 x 128].
// For K == 128 the wave preloads its full B operand set into registers once
// (B frags depend only on wave/lane/k) and reuses it across all row tiles:
// inner loop is pure ds_load + v_wmma, no global traffic, no per-step waits on W.
template <int K, bool DUAL>
__global__ __launch_bounds__(256)
void gemm_wmma_kernel(const float* __restrict__ A1, const float* __restrict__ rowscale,
                      const float* __restrict__ A2,
                      const float* __restrict__ W1, const float* __restrict__ W2,
                      const float* __restrict__ bias, float* __restrict__ C,
                      int nrows, int doAct) {
    constexpr int NK = K / 4;
    constexpr bool PRE = (K == 128);          // preload B into registers
    __shared__ float As1[16 * (K + 1)];       // pad stride K+1 (odd -> bank-clean)
    __shared__ float As2[DUAL ? 16 * 129 : 1];

    const int tid  = threadIdx.x;
    const int wave = tid >> 5;                // 0..7 -> 16-col tile
    const int lane = tid & 31;
    const int lrow = lane & 15;
    const int g    = lane >> 4;               // lane-half
    const int col  = (wave << 4) + lrow;      // output column 0..127
    constexpr int sA = K + 1;
    const int tiles = (nrows + 15) >> 4;

    v2f breg1[PRE ? NK : 1];
    v2f breg2[(PRE && DUAL) ? NK : 1];
    if constexpr (PRE) {
#pragma unroll
        for (int t4 = 0; t4 < NK; ++t4) {
            const int ka = t4 * 4 + 2 * g;    // ISA f32 frag: K = k0 + j + 2g
            breg1[t4].x = W1[(size_t)ka * DD + col];
            breg1[t4].y = W1[(size_t)(ka + 1) * DD + col];
            if constexpr (DUAL) {
                breg2[t4].x = W2[(size_t)ka * DD + col];
                breg2[t4].y = W2[(size_t)(ka + 1) * DD + col];
            }
        }
    }

    for (int t = blockIdx.x; t < tiles; t += gridDim.x) {
        const int row0 = t << 4;
        // stage A tile(s): coalesced global -> LDS, zero-pad tail rows, fold row scaling
        for (int idx = tid; idx < 16 * K; idx += blockDim.x) {
            int r = idx / K, k = idx - r * K;
            int gr = row0 + r;
            float v1 = 0.f, v2 = 0.f;
            if (gr < nrows) {
                v1 = A1[(size_t)gr * K + k];
                if (rowscale) v1 *= rowscale[gr];
                if constexpr (DUAL) v2 = A2[(size_t)gr * DD + k];
            }
            As1[r * sA + k] = v1;
            if constexpr (DUAL) As2[r * 129 + k] = v2;
        }
        __syncthreads();

        v8f acc = {};
#pragma unroll
        for (int t4 = 0; t4 < NK; ++t4) {
            const int ka = t4 * 4 + 2 * g;
            v2f a, b;
            a.x = As1[lrow * sA + ka];
            a.y = As1[lrow * sA + ka + 1];
            if constexpr (PRE) {
                b = breg1[t4];
            } else {
                b.x = W1[(size_t)ka * DD + col];
                b.y = W1[(size_t)(ka + 1) * DD + col];
            }
            acc = __builtin_amdgcn_wmma_f32_16x16x4_f32(false, a, false, b,
                                                        (short)0, acc, false, false);
            if constexpr (DUAL) {
                v2f a2, b2;
                a2.x = As2[lrow * 129 + ka];
                a2.y = As2[lrow * 129 + ka + 1];
                b2 = breg2[t4];
                acc = __builtin_amdgcn_wmma_f32_16x16x4_f32(false, a2, false, b2,
                                                            (short)0, acc, false, false);
            }
        }

        const float bv = bias ? bias[col] : 0.f;
#pragma unroll
        for (int v = 0; v < 8; ++v) {         // C frag: row = v + 8g, col = lane&15
            int r = row0 + v + 8 * g;
            if (r < nrows) {
                float x = acc[v] + bv;
                if (doAct) x = leakyf(x);
                C[(size_t)r * DD + col] = x;
            }
        }
        __syncthreads();
    }
}

__global__ void count_kernel(float* __restrict__ cnt, const int* __restrict__ batch, int n) {
    int i = blockIdx.x * blockDim.x + threadIdx.x;
    if (i < n) atomicAdd(&cnt[batch[i]], 1.0f);
}

__global__ void pool_kernel(const float* __restrict__ h, const int* __restrict__ batch,
                            float* __restrict__ gmax, float* __restrict__ gsum, int n) {
    long tid = (long)blockIdx.x * blockDim.x + threadIdx.x;
    if (tid >= (long)n * 32) return;
    int r = (int)(tid >> 5);
    int d0 = ((int)tid & 31) * 4;
    int b = batch[r];
#pragma unroll
    for (int i = 0; i < 4; ++i) {
        float v = h[(size_t)r * DD + d0 + i];
        atomicAdd(&gsum[(size_t)b * DD + d0 + i], v);
        atomicMaxFloat(&gmax[(size_t)b * DD + d0 + i], v);
    }
}

__global__ void buildp_kernel(const float* __restrict__ gmax, const float* __restrict__ gsum,
                              const float* __restrict__ cnt, float* __restrict__ p) {
    int tid = blockIdx.x * blockDim.x + threadIdx.x;
    if (tid >= BB * 3 * DD) return;
    int b = tid / (3 * DD), j = tid % (3 * DD);
    float c = cnt[b], v;
    if (j < DD)            v = (c > 0.f) ? gmax[b * DD + j] : 0.f;
    else if (j < 2 * DD)   v = gsum[b * DD + (j - DD)] / fmaxf(c, 1.f);
    else                   v = gsum[b * DD + (j - 2 * DD)];
    p[tid] = v;
}

__global__ void final_kernel(const float* __restrict__ t2, const float* __restrict__ Wf3,
                             const float* __restrict__ bf3, float* __restrict__ out, int nb) {
    int b = blockIdx.x * blockDim.x + threadIdx.x;
    if (b >= nb) return;
    float s = bf3[0];
#pragma unroll 8
    for (int k = 0; k < DD; ++k) s += t2[(size_t)b * DD + k] * Wf3[k];
    out[b] = s;
}

extern "C" void kernel_launch(void* const* d_in, const int* in_sizes, int n_in,
                              void* d_out, int out_size, void* d_ws, size_t ws_size,
                              hipStream_t stream) {
    const float* x     = (const float*)d_in[0];
    const int*   ei    = (const int*)d_in[1];
    const int*   src   = ei;
    const int*   dst   = ei + EE;
    const int*   batch = (const int*)d_in[2];
    const float* W0  = (const float*)d_in[3];  const float* b0  = (const float*)d_in[4];
    const float* Wl  = (const float*)d_in[5];  const float* bl  = (const float*)d_in[6];
    const float* Wr  = (const float*)d_in[7];
    const float* Wrel= (const float*)d_in[8];  const float* brel= (const float*)d_in[9];
    const float* Wroot=(const float*)d_in[10];
    const float* W3  = (const float*)d_in[11]; const float* b3  = (const float*)d_in[12];
    const float* Wf1 = (const float*)d_in[13]; const float* bf1 = (const float*)d_in[14];
    const float* Wf2 = (const float*)d_in[15]; const float* bf2 = (const float*)d_in[16];
    const float* Wf3 = (const float*)d_in[17]; const float* bf3 = (const float*)d_in[18];
    const float* g1  = (const float*)d_in[19]; const float* be1 = (const float*)d_in[20];
    const float* g2  = (const float*)d_in[21]; const float* be2 = (const float*)d_in[22];
    const float* g3  = (const float*)d_in[23]; const float* be3 = (const float*)d_in[24];

    float* ws   = (float*)d_ws;
    float* buf0 = ws;                                // N*128
    float* buf1 = buf0 + (size_t)NN * DD;            // N*128
    float* buf2 = buf1 + (size_t)NN * DD;            // N*128
    float* deg  = buf2 + (size_t)NN * DD;            // N
    float* rs   = deg + NN;                          // N
    float* stats= rs + NN;                           // 256

    auto GS = [](long n, int b) { return (int)((n + b - 1) / b); };
    const int T = 256;
    const long ND = (long)NN * DD;
    const int gtiles = (NN + 15) >> 4;               // 3125

    // degrees
    fill_kernel<<<GS(NN, T), T, 0, stream>>>(deg, NN, 0.f);
    deg_kernel<<<GS(EE, T), T, 0, stream>>>(deg, dst, EE);
    rs_kernel<<<GS(NN, T), T, 0, stream>>>(rs, deg, NN);

    // layer 1: GCN(x, W0, b0) -> leaky -> bn1         (h in buf2)
    lin0_kernel<<<GS(ND, T), T, 0, stream>>>(x, W0, buf0, NN);
    fill_kernel<<<GS(ND, T), T, 0, stream>>>(buf1, ND, 0.f);
    scatter_kernel<<<GS((long)EE * 32, T), T, 0, stream>>>(buf1, buf0, src, dst, deg, EE, 1);
    gcn_finish_kernel<<<GS(ND, T), T, 0, stream>>>(buf1, buf0, deg, b0, buf2, NN);
    fill_kernel<<<1, T, 0, stream>>>(stats, 2 * DD, 0.f);
    bn_stats_kernel<<<512, T, 0, stream>>>(buf2, stats, NN);
    bn_apply_kernel<<<GS(ND, T), T, 0, stream>>>(buf2, stats, g1, be1, NN);

    // layer 2: SAGE  mean_nb@Wl + h@Wr + bl -> leaky -> bn2   (h in buf0)
    fill_kernel<<<GS(ND, T), T, 0, stream>>>(buf1, ND, 0.f);
    scatter_kernel<<<GS((long)EE * 32, T), T, 0, stream>>>(buf1, buf2, src, dst, deg, EE, 0);
    gemm_wmma_kernel<128, true><<<gtiles, T, 0, stream>>>(buf1, rs, buf2, Wl, Wr, bl, buf0, NN, 1);
    fill_kernel<<<1, T, 0, stream>>>(stats, 2 * DD, 0.f);
    bn_stats_kernel<<<512, T, 0, stream>>>(buf0, stats, NN);
    bn_apply_kernel<<<GS(ND, T), T, 0, stream>>>(buf0, stats, g2, be2, NN);

    // layer 3: GraphConv  agg@Wrel + h@Wroot + brel -> leaky  (h in buf2)
    fill_kernel<<<GS(ND, T), T, 0, stream>>>(buf1, ND, 0.f);
    scatter_kernel<<<GS((long)EE * 32, T), T, 0, stream>>>(buf1, buf0, src, dst, deg, EE, 0);
    gemm_wmma_kernel<128, true><<<gtiles, T, 0, stream>>>(buf1, nullptr, buf0, Wrel, Wroot, brel, buf2, NN, 1);

    // layer 4: GCN(h, W3, b3) -> leaky                (h4 in buf2)
    gemm_wmma_kernel<128, false><<<gtiles, T, 0, stream>>>(buf2, nullptr, nullptr, W3, nullptr, nullptr, buf0, NN, 0);
    fill_kernel<<<GS(ND, T), T, 0, stream>>>(buf1, ND, 0.f);
    scatter_kernel<<<GS((long)EE * 32, T), T, 0, stream>>>(buf1, buf0, src, dst, deg, EE, 1);
    gcn_finish_kernel<<<GS(ND, T), T, 0, stream>>>(buf1, buf0, deg, b3, buf2, NN);

    // global pooling (temps alias freed buf0/buf1)
    float* gmax = buf0;
    float* gsum = buf0 + (size_t)BB * DD;
    float* cnt  = buf0 + (size_t)2 * BB * DD;
    fill_kernel<<<GS((long)BB * DD, T), T, 0, stream>>>(gmax, (long)BB * DD, -3.4e38f);
    fill_kernel<<<GS((long)BB * DD, T), T, 0, stream>>>(gsum, (long)BB * DD, 0.f);
    fill_kernel<<<1, T, 0, stream>>>(cnt, BB, 0.f);
    count_kernel<<<GS(NN, T), T, 0, stream>>>(cnt, batch, NN);
    pool_kernel<<<GS((long)NN * 32, T), T, 0, stream>>>(buf2, batch, gmax, gsum, NN);

    float* p  = buf1;
    float* t1 = p + (size_t)BB * 3 * DD;
    float* t2 = t1 + (size_t)BB * DD;
    buildp_kernel<<<GS((long)BB * 3 * DD, T), T, 0, stream>>>(gmax, gsum, cnt, p);

    // MLP head (B=512 rows -> 32 tiles)
    const int btiles = (BB + 15) >> 4;
    gemm_wmma_kernel<384, false><<<btiles, T, 0, stream>>>(p, nullptr, nullptr, Wf1, nullptr, bf1, t1, BB, 1);
    gemm_wmma_kernel<128, false><<<btiles, T, 0, stream>>>(t1, nullptr, nullptr, Wf2, nullptr, bf2, t2, BB, 1);
    fill_kernel<<<1, T, 0, stream>>>(stats, 2 * DD, 0.f);
    bn_stats_kernel<<<64, T, 0, stream>>>(t2, stats, BB);
    bn_apply_kernel<<<GS((long)BB * DD, T), T, 0, stream>>>(t2, stats, g3, be3, BB);
    final_kernel<<<GS(BB, T), T, 0, stream>>>(t2, Wf3, bf3, (float*)d_out, BB);
}